// IGA_13881334300908
// MI455X (gfx1250) — compile-verified
//
#include <hip/hip_runtime.h>
#include <math.h>
#include <stdint.h>

typedef __attribute__((ext_vector_type(2))) float v2f;
typedef __attribute__((ext_vector_type(8))) float v8f;

#define HW     16384
#define WIDTH  128
#define CIN    192
#define C3     576
#define CPH    48
#define BATCH  8

static __device__ __forceinline__ v8f wmma4(v2f a, v2f b, v8f c) {
  // D = A(16x4 f32) x B(4x16 f32) + C(16x16 f32)
  return __builtin_amdgcn_wmma_f32_16x16x4_f32(false, a, false, b, (short)0, c,
                                               false, false);
}

// ---------------------------------------------------------------------------
// 1x1 conv as GEMM: C[b,m,n] = sum_k W[m,k] * X[b,k,n],  N = 16384
// block = 128 threads (4 waves); block tile = 64M x 64N.
// X tile staged into LDS with GLOBAL_LOAD_ASYNC_TO_LDS_B128 (ASYNCcnt path):
// 16B per lane go straight to LDS without a VGPR round-trip.
// ---------------------------------------------------------------------------
template <int M, int K>
__global__ __launch_bounds__(128) void gemm1x1(const float* __restrict__ W,
                                               const float* __restrict__ X,
                                               float* __restrict__ C) {
  constexpr int LS = 72;  // padded LDS row stride (2*72 mod 64 = 16 -> halves on disjoint banks)
  __shared__ float xs[K * LS];
  const int b = blockIdx.z;
  const int n0 = blockIdx.x * 64;
  const int m_base = blockIdx.y * 64;
  const int tid = threadIdx.x;

  const float* Xb = X + (size_t)b * K * HW + n0;
  {
    // Each pass: 128 lanes x 16B = 8 rows of 64 floats. 24 passes for K=192.
    const int c4 = (tid & 15) * 4;  // column group within the 64-wide tile
    const int r0 = tid >> 4;        // 0..7
    for (int r = r0; r < K; r += 8) {
      const uint32_t lds_b = (uint32_t)(uintptr_t)&xs[r * LS + c4];
      const float* gp = Xb + (size_t)r * HW + c4;
      asm volatile("global_load_async_to_lds_b128 %0, %1, off"
                   :
                   : "v"(lds_b), "v"(gp)
                   : "memory");
    }
    asm volatile("s_wait_asynccnt 0x0" ::: "memory");
  }
  __syncthreads();

  const int lane = tid & 31;
  const int lm = lane & 15, lh = lane >> 4;
  const int m0 = m_base + (tid >> 5) * 16;
  const float* Wr = W + (size_t)(m0 + lm) * K;

  v8f c0 = {}, c1 = {}, c2 = {}, c3 = {};
#pragma unroll 4
  for (int k0 = 0; k0 < K; k0 += 4) {
    v2f a;
    a.x = Wr[k0 + 2 * lh];
    a.y = Wr[k0 + 2 * lh + 1];
    const float* x0 = &xs[(k0 + 2 * lh) * LS + lm];
    v2f b0, b1, b2, b3;
    b0.x = x0[0];  b0.y = x0[LS];
    b1.x = x0[16]; b1.y = x0[16 + LS];
    b2.x = x0[32]; b2.y = x0[32 + LS];
    b3.x = x0[48]; b3.y = x0[48 + LS];
    c0 = wmma4(a, b0, c0);
    c1 = wmma4(a, b1, c1);
    c2 = wmma4(a, b2, c2);
    c3 = wmma4(a, b3, c3);
  }

  float* Cb = C + (size_t)b * M * HW + n0 + lm;
#pragma unroll
  for (int r = 0; r < 8; ++r) {
    const size_t row = (size_t)(m0 + r + 8 * lh);
    float* cr = Cb + row * HW;
    cr[0] = c0[r];
    cr[16] = c1[r];
    cr[32] = c2[r];
    cr[48] = c3[r];
  }
}

// ---------------------------------------------------------------------------
// depthwise 3x3, pad 1 (memory-bound; plain VALU)
// ---------------------------------------------------------------------------
__global__ __launch_bounds__(256) void dwconv3x3(const float* __restrict__ in,
                                                 const float* __restrict__ w,
                                                 float* __restrict__ out) {
  const size_t idx = (size_t)blockIdx.x * 256 + threadIdx.x;  // b*576*HW + ch*HW + p
  const int p = (int)(idx & (HW - 1));
  const size_t ch_b = idx >> 14;  // b*576 + ch
  const int ch = (int)(ch_b % C3);
  const int x = p & (WIDTH - 1), y = p >> 7;
  const float* wp = w + ch * 9;
  const float* ip = in + (ch_b << 14);
  float acc = 0.f;
#pragma unroll
  for (int dy = -1; dy <= 1; ++dy) {
    const int yy = y + dy;
    if (yy < 0 || yy >= WIDTH) continue;
#pragma unroll
    for (int dx = -1; dx <= 1; ++dx) {
      const int xx = x + dx;
      if (xx < 0 || xx >= WIDTH) continue;
      acc += wp[(dy + 1) * 3 + (dx + 1)] * ip[yy * WIDTH + xx];
    }
  }
  out[idx] = acc;
}

// ---------------------------------------------------------------------------
// Stage A of attention: partial row-norms of q,k and partial q.k^T logits.
// grid = (64 K-chunks of 256, 32 bh), block = 256 (8 waves).
// Each wave: 32-wide K-slice, 9 WMMA tiles accumulated; LDS ds_add_f32
// reduction across waves; global f32 atomics into attn_raw.
// ---------------------------------------------------------------------------
__global__ __launch_bounds__(256) void attn_qk_partial(const float* __restrict__ qkv,
                                                       float* __restrict__ norms,
                                                       float* __restrict__ attn_raw) {
  const int bh = blockIdx.y;
  const int b = bh >> 2, h = bh & 3;
  const int kbase = blockIdx.x * 256;
  const int tid = threadIdx.x, wave = tid >> 5, lane = tid & 31;

  const float* q = qkv + ((size_t)b * C3 + h * CPH) * HW;
  const float* kmat = q + (size_t)CIN * HW;

  // --- norm partials: 96 jobs (48 q rows + 48 k rows), 12 per wave
  for (int j = wave; j < 96; j += 8) {
    const float* row =
        (j < 48 ? q + (size_t)j * HW : kmat + (size_t)(j - 48) * HW) + kbase;
    float s = 0.f;
#pragma unroll
    for (int ii = 0; ii < 8; ++ii) {
      const float vv = row[lane + 32 * ii];
      s += vv * vv;
    }
#pragma unroll
    for (int off = 16; off; off >>= 1) s += __shfl_xor(s, off, 32);
    if (lane == 0) atomicAdd(&norms[bh * 96 + j], s);
  }

  // --- q.k^T partial (M=48, N=48, K-slice of 32 per wave)
  const int lm = lane & 15, lh = lane >> 4;
  v8f c[3][3] = {};
  const int kw = kbase + wave * 32;
  for (int st = 0; st < 8; ++st) {
    const int k0 = kw + st * 4;
    v2f a[3], bt[3];
#pragma unroll
    for (int t = 0; t < 3; ++t) {
      const float* qr = q + (size_t)(t * 16 + lm) * HW + k0 + 2 * lh;
      a[t].x = qr[0];
      a[t].y = qr[1];
      const float* kr = kmat + (size_t)(t * 16 + lm) * HW + k0 + 2 * lh;
      bt[t].x = kr[0];
      bt[t].y = kr[1];
    }
#pragma unroll
    for (int mt = 0; mt < 3; ++mt)
#pragma unroll
      for (int nt = 0; nt < 3; ++nt) c[mt][nt] = wmma4(a[mt], bt[nt], c[mt][nt]);
  }

  __shared__ float s_attn[48 * 48];
  for (int i = tid; i < 48 * 48; i += 256) s_attn[i] = 0.f;
  __syncthreads();
#pragma unroll
  for (int mt = 0; mt < 3; ++mt)
#pragma unroll
    for (int nt = 0; nt < 3; ++nt)
#pragma unroll
      for (int r = 0; r < 8; ++r)
        atomicAdd(&s_attn[(mt * 16 + r + 8 * lh) * 48 + nt * 16 + lm], c[mt][nt][r]);
  __syncthreads();

  float* araw = attn_raw + (size_t)bh * 2304;
  for (int i = tid; i < 2304; i += 256) atomicAdd(&araw[i], s_attn[i]);
}

// ---------------------------------------------------------------------------
// Stage B: finalize norms, scale by temperature, row softmax (48x48 per bh).
// ---------------------------------------------------------------------------
__global__ __launch_bounds__(64) void attn_softmax(const float* __restrict__ norms,
                                                   const float* __restrict__ attn_raw,
                                                   const float* __restrict__ temperature,
                                                   float* __restrict__ attn_prob) {
  const int bh = blockIdx.x;
  const int h = bh & 3;
  const int t = threadIdx.x;
  __shared__ float rq[48], rk[48];
  if (t < 48) {
    rq[t] = 1.f / fmaxf(sqrtf(norms[bh * 96 + t]), 1e-12f);
    rk[t] = 1.f / fmaxf(sqrtf(norms[bh * 96 + 48 + t]), 1e-12f);
  }
  __syncthreads();
  if (t < 48) {
    const float temp = temperature[h];
    const float* ar = attn_raw + (size_t)bh * 2304 + t * 48;
    float vals[48];
    float mx = -INFINITY;
#pragma unroll
    for (int d = 0; d < 48; ++d) {
      vals[d] = temp * rq[t] * rk[d] * ar[d];
      mx = fmaxf(mx, vals[d]);
    }
    float s = 0.f;
#pragma unroll
    for (int d = 0; d < 48; ++d) {
      vals[d] = __expf(vals[d] - mx);
      s += vals[d];
    }
    const float inv = 1.f / s;
    float* ap = attn_prob + (size_t)bh * 2304 + t * 48;
#pragma unroll
    for (int d = 0; d < 48; ++d) ap[d] = vals[d] * inv;
  }
}

// ---------------------------------------------------------------------------
// Stage C: out = attn(48x48) @ v(48xHW).  grid = (8 N-chunks, 32 bh).
// attn staged in LDS; all 36 A-fragments preloaded into registers.
// ---------------------------------------------------------------------------
__global__ __launch_bounds__(256) void attn_av(const float* __restrict__ attn_prob,
                                               const float* __restrict__ qkv,
                                               float* __restrict__ y) {
  const int bh = blockIdx.y;
  const int b = bh >> 2, h = bh & 3;
  const int tid = threadIdx.x, wave = tid >> 5, lane = tid & 31;
  const int lm = lane & 15, lh = lane >> 4;

  constexpr int AS = 52;  // padded row stride
  __shared__ float s_a[48 * AS];
  const float* ap = attn_prob + (size_t)bh * 2304;
  for (int i = tid; i < 2304; i += 256) s_a[(i / 48) * AS + (i % 48)] = ap[i];
  __syncthreads();

  v2f af[3][12];
#pragma unroll
  for (int mt = 0; mt < 3; ++mt)
#pragma unroll
    for (int ks = 0; ks < 12; ++ks) {
      const float* r = &s_a[(mt * 16 + lm) * AS + ks * 4 + 2 * lh];
      af[mt][ks].x = r[0];
      af[mt][ks].y = r[1];
    }

  const float* v = qkv + ((size_t)b * C3 + 2 * CIN + h * CPH) * HW;
  float* yb = y + ((size_t)b * CIN + h * CPH) * HW;
  const int nwave = blockIdx.x * 2048 + wave * 256;

  for (int t = 0; t < 16; ++t) {
    const int col = nwave + t * 16 + lm;
    v8f c0 = {}, c1 = {}, c2 = {};
#pragma unroll
    for (int ks = 0; ks < 12; ++ks) {
      v2f bt;
      const float* vr = v + (size_t)(ks * 4 + 2 * lh) * HW + col;
      bt.x = vr[0];
      bt.y = vr[HW];
      c0 = wmma4(af[0][ks], bt, c0);
      c1 = wmma4(af[1][ks], bt, c1);
      c2 = wmma4(af[2][ks], bt, c2);
    }
#pragma unroll
    for (int r = 0; r < 8; ++r) {
      const int rr = r + 8 * lh;
      yb[(size_t)rr * HW + col] = c0[r];
      yb[(size_t)(16 + rr) * HW + col] = c1[r];
      yb[(size_t)(32 + rr) * HW + col] = c2[r];
    }
  }
}

// ---------------------------------------------------------------------------
extern "C" void kernel_launch(void* const* d_in, const int* in_sizes, int n_in,
                              void* d_out, int out_size, void* d_ws, size_t ws_size,
                              hipStream_t stream) {
  const float* x = (const float*)d_in[0];
  const float* w_in = (const float*)d_in[1];
  const float* w_dw = (const float*)d_in[2];
  const float* temperature = (const float*)d_in[3];
  const float* w_out = (const float*)d_in[4];
  float* out = (float*)d_out;

  const size_t SZ_QKV = (size_t)BATCH * C3 * HW;  // elements
  float* buf1 = (float*)d_ws;         // qkv pre-dw; later reused as attention output y
  float* buf2 = buf1 + SZ_QKV;        // qkv post-dw
  float* norms = buf2 + SZ_QKV;       // 32 * 96
  float* attn_raw = norms + 32 * 96;  // 32 * 2304
  float* attn_prob = attn_raw + 32 * 2304;

  hipMemsetAsync((void*)norms, 0, (size_t)(32 * 96 + 32 * 2304) * sizeof(float),
                 stream);

  // 1) conv_in (1x1): qkv = w_in @ x
  gemm1x1<C3, CIN><<<dim3(HW / 64, C3 / 64, BATCH), 128, 0, stream>>>(w_in, x, buf1);
  // 2) depthwise 3x3
  dwconv3x3<<<(BATCH * C3 * HW) / 256, 256, 0, stream>>>(buf1, w_dw, buf2);
  // 3a) norms + q.k^T partial logits
  attn_qk_partial<<<dim3(64, 32), 256, 0, stream>>>(buf2, norms, attn_raw);
  // 3b) scale + softmax
  attn_softmax<<<32, 64, 0, stream>>>(norms, attn_raw, temperature, attn_prob);
  // 3c) out = attn @ v   (writes into buf1, which is free now)
  attn_av<<<dim3(8, 32), 256, 0, stream>>>(attn_prob, buf2, buf1);
  // 4) conv_out (1x1)
  gemm1x1<CIN, CIN><<<dim3(HW / 64, CIN / 64, BATCH), 128, 0, stream>>>(w_out, buf1, out);
}